// _Attention_74723841015948
// MI455X (gfx1250) — compile-verified
//
#include <hip/hip_runtime.h>
#include <hip/hip_bf16.h>

// Problem constants (from reference)
#define BB    256
#define TT    2048
#define RNN   1024
#define ENC   512
#define HID   128
#define NF    32
#define KS    31
#define PADL  15
#define TP    (TT + 2 * PADL)   // 2078 padded attention-weight row

typedef __attribute__((ext_vector_type(2))) float f32x2;
typedef __attribute__((ext_vector_type(8))) float f32x8;

// V_WMMA_F32_16X16X4_F32: D = A(16x4 f32) * B(4x16 f32) + C(16x16 f32)
// 8-arg form: (neg_a, A, neg_b, B, c_mod, C, reuse_a, reuse_b)
__device__ __forceinline__ f32x8 wmma_f32(f32x2 a, f32x2 b, f32x8 c) {
    return __builtin_amdgcn_wmma_f32_16x16x4_f32(false, a, false, b, (short)0, c,
                                                 false, false);
}

// ---------------------------------------------------------------------------
// Kernel 1: pq = query[256x1024] @ Wq[1024x128] via WMMA f32 16x16x4.
// 128 output tiles of 16x16; one wave per tile; 256 K-steps of 4.
// ---------------------------------------------------------------------------
__global__ __launch_bounds__(256)
void pq_gemm_wmma(const float* __restrict__ q, const float* __restrict__ wq,
                  float* __restrict__ pq) {
    const int tid  = threadIdx.x;
    const int wave = tid >> 5;
    const int lane = tid & 31;
    const int half = lane >> 4;
    const int l    = lane & 15;

    const int wg = blockIdx.x * 8 + wave;   // 0..127 global wave id
    const int m0 = (wg >> 3) * 16;          // batch-row tile
    const int n0 = (wg & 7) * 16;           // hid tile

    const int m = m0 + l;                   // A row for this lane (both halves)
    f32x8 c = {};
    for (int k = 0; k < RNN; k += 4) {
        const int ka = k + 2 * half;        // lanes 0-15: K0/K1, lanes 16-31: K2/K3
        f32x2 a, b;
        a.x = q[(size_t)m * RNN + ka];
        a.y = q[(size_t)m * RNN + ka + 1];
        b.x = wq[(size_t)ka * HID + n0 + l];
        b.y = wq[(size_t)(ka + 1) * HID + n0 + l];
        c = wmma_f32(a, b, c);
    }
#pragma unroll
    for (int r = 0; r < 8; ++r) {           // C: VGPR r -> M=r (lanes<16) / M=r+8
        const int mm = m0 + r + half * 8;
        pq[(size_t)mm * HID + n0 + l] = c[r];
    }
}

// ---------------------------------------------------------------------------
// Kernel 2: fused conv + ploc(WMMA) + tanh-energies + softmax + context.
// One 256-thread block (8 wave32) per batch row b.
// ---------------------------------------------------------------------------
__global__ __launch_bounds__(256, 2)
void attn_fused(const float* __restrict__ q,      const float* __restrict__ memory,
                const float* __restrict__ pmem,   const float* __restrict__ awc,
                const unsigned char* __restrict__ mask,
                const float* __restrict__ wq,     const float* __restrict__ conv_w,
                const float* __restrict__ wloc,   const float* __restrict__ vvec,
                const float* __restrict__ pq_g,   // may be null -> compute in-block
                float* __restrict__ out_ctx, float* __restrict__ out_w) {
    __shared__ float s_aw[2][TP];          // padded attention weights     16.6 KB
    __shared__ float s_wc[62][NF];         // conv weights [c*31+k][f]      7.9 KB
    __shared__ float s_wl[NF][HID];        // Wloc                         16.0 KB
    __shared__ float s_pq[HID];
    __shared__ float s_v[HID];
    __shared__ float s_loc[8][16][33];     // per-wave conv tile (padded)  16.9 KB
    __shared__ float s_e[TT];              // energies -> weights           8.0 KB
    __shared__ float s_red[8];

    const int b    = blockIdx.x;
    const int tid  = threadIdx.x;
    const int wave = tid >> 5;
    const int lane = tid & 31;
    const int half = lane >> 4;
    const int l    = lane & 15;

    // ---------------- Phase 0: stage operands into LDS ----------------
    for (int i = tid; i < 2 * TP; i += 256) {
        const int ch = i / TP, j = i - ch * TP;
        const int t = j - PADL;
        float val = 0.f;
        if (t >= 0 && t < TT)
            val = awc[((size_t)b * 2 + ch) * TT + t];
        s_aw[ch][j] = val;
    }
    for (int i = tid; i < 62 * NF; i += 256) {
        const int ck = i >> 5, f = i & 31;          // ck = c*31+k
        s_wc[ck][f] = conv_w[f * 62 + ck];
    }
    for (int i = tid; i < NF * HID; i += 256)
        s_wl[i >> 7][i & 127] = wloc[i];
    if (tid < HID) {
        s_v[tid] = vvec[tid];
        if (pq_g) {
            s_pq[tid] = pq_g[(size_t)b * HID + tid];
        } else {
            float s = 0.f;
            for (int k = 0; k < RNN; ++k)
                s += q[(size_t)b * RNN + k] * wq[(size_t)k * HID + tid];
            s_pq[tid] = s;
        }
    }
    __syncthreads();

    // ---------------- Phase 1: per-wave 16-t tiles --------------------
    // conv (VALU) -> s_loc, then ploc = loc @ Wloc via WMMA f32 16x16x4,
    // fold pq + pmem + tanh + v-dot on the accumulator fragments.
    for (int tile = wave; tile < TT / 16; tile += 8) {
        const int t0 = tile * 16;

        // conv: lane == filter f, 16 time steps, K = 2 ch x 31 taps
        {
            const int f = lane;
            for (int ttl = 0; ttl < 16; ++ttl) {
                float s = 0.f;
#pragma unroll
                for (int ch = 0; ch < 2; ++ch)
#pragma unroll
                    for (int k = 0; k < KS; ++k)
                        s += s_wc[ch * KS + k][f] * s_aw[ch][t0 + ttl + k];
                s_loc[wave][ttl][f] = s;
            }
        }
        __builtin_amdgcn_wave_barrier();   // cross-lane LDS visibility (same wave)

        float acc[8];
#pragma unroll
        for (int r = 0; r < 8; ++r) acc[r] = 0.f;

        for (int nc = 0; nc < HID / 16; ++nc) {        // 8 hid chunks of 16
            f32x8 c = {};
#pragma unroll
            for (int kk = 0; kk < 8; ++kk) {           // K=32 in steps of 4
                const int kb = kk * 4 + 2 * half;
                f32x2 a, bfr;
                a.x   = s_loc[wave][l][kb];
                a.y   = s_loc[wave][l][kb + 1];
                bfr.x = s_wl[kb][nc * 16 + l];
                bfr.y = s_wl[kb + 1][nc * 16 + l];
                c = wmma_f32(a, bfr, c);
            }
            const int n = nc * 16 + l;
            const float pqn = s_pq[n];
            const float vn  = s_v[n];
            const float* pm =
                pmem + ((size_t)b * TT + t0 + half * 8) * HID + n;
#pragma unroll
            for (int r = 0; r < 8; ++r) {              // element (M=r+half*8, N=n)
                const float pmv = __builtin_nontemporal_load(pm + (size_t)r * HID);
                acc[r] += vn * tanhf(pqn + c[r] + pmv);
            }
        }
        // reduce over the 16 lanes sharing each time step
#pragma unroll
        for (int r = 0; r < 8; ++r) {
            float x = acc[r];
            x += __shfl_xor(x, 1, 32);
            x += __shfl_xor(x, 2, 32);
            x += __shfl_xor(x, 4, 32);
            x += __shfl_xor(x, 8, 32);
            if (l == 0) s_e[t0 + half * 8 + r] = x;
        }
        __builtin_amdgcn_wave_barrier();
    }
    __syncthreads();

    // ---------------- Phase 2: masked softmax over T ------------------
    float lmax = -3.4e38f;
    for (int t = tid; t < TT; t += 256) {
        float e = s_e[t];
        if (mask[(size_t)b * TT + t]) e = -3.4e38f;
        s_e[t] = e;
        lmax = fmaxf(lmax, e);
    }
#pragma unroll
    for (int off = 1; off < 32; off <<= 1)
        lmax = fmaxf(lmax, __shfl_xor(lmax, off, 32));
    if (lane == 0) s_red[wave] = lmax;
    __syncthreads();
    float bmax = s_red[0];
#pragma unroll
    for (int w2 = 1; w2 < 8; ++w2) bmax = fmaxf(bmax, s_red[w2]);
    __syncthreads();

    float lsum = 0.f;
    for (int t = tid; t < TT; t += 256) {
        const float e = __expf(s_e[t] - bmax);
        s_e[t] = e;
        lsum += e;
    }
#pragma unroll
    for (int off = 1; off < 32; off <<= 1)
        lsum += __shfl_xor(lsum, off, 32);
    if (lane == 0) s_red[wave] = lsum;
    __syncthreads();
    float bsum = 0.f;
#pragma unroll
    for (int w2 = 0; w2 < 8; ++w2) bsum += s_red[w2];
    const float inv = 1.0f / bsum;
    for (int t = tid; t < TT; t += 256) {
        const float wgt = s_e[t] * inv;
        s_e[t] = wgt;
        out_w[(size_t)b * TT + t] = wgt;
    }
    __syncthreads();

    // ---------------- Phase 3: context = attn @ memory ----------------
    // 256 threads x float2 == one full 512-float row per step, coalesced.
    f32x2 ctx = {0.f, 0.f};
    const f32x2* mb2 = (const f32x2*)(memory + (size_t)b * TT * ENC);
    for (int t = 0; t < TT; t += 4) {
        __builtin_prefetch(mb2 + (size_t)(t + 16) * (ENC / 2) + tid, 0, 1);
#pragma unroll
        for (int u = 0; u < 4; ++u) {
            const float wgt = s_e[t + u];
            const f32x2 m2 =
                __builtin_nontemporal_load(mb2 + (size_t)(t + u) * (ENC / 2) + tid);
            ctx.x += wgt * m2.x;
            ctx.y += wgt * m2.y;
        }
    }
    ((f32x2*)out_ctx)[(size_t)b * (ENC / 2) + tid] = ctx;
}

// ---------------------------------------------------------------------------
extern "C" void kernel_launch(void* const* d_in, const int* in_sizes, int n_in,
                              void* d_out, int out_size, void* d_ws, size_t ws_size,
                              hipStream_t stream) {
    const float*         q    = (const float*)d_in[0];          // [B, RNN]
    const float*         mem  = (const float*)d_in[1];          // [B, T, ENC]
    const float*         pmem = (const float*)d_in[2];          // [B, T, HID]
    const float*         awc  = (const float*)d_in[3];          // [B, 2, T]
    const unsigned char* mask = (const unsigned char*)d_in[4];  // [B, T] bool
    const float*         wq   = (const float*)d_in[5];          // [RNN, HID]
    const float*         cw   = (const float*)d_in[6];          // [NF, 2, KS]
    const float*         wl   = (const float*)d_in[7];          // [NF, HID]
    const float*         vv   = (const float*)d_in[8];          // [HID]

    float* out     = (float*)d_out;
    float* out_ctx = out;                       // [B, ENC]
    float* out_w   = out + (size_t)BB * ENC;    // [B, T]

    float* pq_ws = nullptr;
    if (ws_size >= (size_t)BB * HID * sizeof(float)) {
        pq_ws = (float*)d_ws;
        pq_gemm_wmma<<<16, 256, 0, stream>>>(q, wq, pq_ws);
    }
    attn_fused<<<BB, 256, 0, stream>>>(q, mem, pmem, awc, mask, wq, cw, wl, vv,
                                       pq_ws, out_ctx, out_w);
}